// ColorLoss_68453188763708
// MI455X (gfx1250) — compile-verified
//
#include <hip/hip_runtime.h>
#include <stdint.h>
#include <stddef.h>

// ---------------- configuration ----------------
#define TILE      1024          // pixels per TDM tile (divides 512*512)
#define THREADS   256           // 8 waves (wave32)
#define HW        262144        // 512*512 channel-plane stride (elements)
#define MAXBLOCKS 2048

typedef unsigned int u32x4 __attribute__((ext_vector_type(4)));
typedef int          i32x8 __attribute__((ext_vector_type(8)));
typedef int          i32x4 __attribute__((ext_vector_type(4)));

#if defined(__HIP_DEVICE_COMPILE__) && defined(__has_builtin)
#if __has_builtin(__builtin_amdgcn_tensor_load_to_lds) && __has_builtin(__builtin_amdgcn_s_wait_tensorcnt)
#define HAVE_TDM 1
#endif
#endif
#ifndef HAVE_TDM
#define HAVE_TDM 0
#endif

// ---------------- fast math on HW transcendental units ----------------
// v_sin/v_cos take input in revolutions (x * 1/(2*pi)); we fold deg->rev.
__device__ __forceinline__ float hsqrt(float x) { return __builtin_amdgcn_sqrtf(x); }
__device__ __forceinline__ float hrcp (float x) { return __builtin_amdgcn_rcpf(x); }
__device__ __forceinline__ float hrsq (float x) { return __builtin_amdgcn_rsqf(x); }
__device__ __forceinline__ float hexp2(float x) { return __builtin_amdgcn_exp2f(x); }
__device__ __forceinline__ float hlog2(float x) { return __builtin_amdgcn_logf(x); }
__device__ __forceinline__ float sin_rev(float r) { return __builtin_amdgcn_sinf(r); }
__device__ __forceinline__ float cosd(float d) { return __builtin_amdgcn_cosf(d * (1.0f/360.0f)); }

// sRGB -> linear (clamped to [0,1]); pow(t,2.4) = exp2(2.4*log2 t) on HW units
__device__ __forceinline__ float s2lin(float c) {
    c = fminf(fmaxf(c, 0.0f), 1.0f);
    float lo = c * (1.0f/12.92f);
    float t  = (c + 0.055f) * (1.0f/1.055f);
    float hi = hexp2(2.4f * hlog2(t));
    return c > 0.04045f ? hi : lo;
}

// CIE f(): cbrt via exp2(log2/3)
__device__ __forceinline__ float labf(float t) {
    float c = hexp2((1.0f/3.0f) * hlog2(t));
    float l = fmaf(7.787f, t, 16.0f/116.0f);
    return t > 0.008856f ? c : l;
}

// D65 white folded into the matrix rows: M[k][c] / WHITE[k]
__device__ __forceinline__ void rgb2lab(float r, float g, float b,
                                        float& L, float& A, float& B) {
    r = s2lin(r); g = s2lin(g); b = s2lin(b);
    float fx = labf(fmaf(0.43394992f, r, fmaf(0.37621495f, g, 0.18982564f * b)));
    float fy = labf(fmaf(0.21267100f, r, fmaf(0.71516000f, g, 0.07216900f * b)));
    float fz = labf(fmaf(0.01775656f, r, fmaf(0.10946892f, g, 0.87271290f * b)));
    L = fmaf(116.0f, fy, -16.0f);
    A = 500.0f * (fx - fy);
    B = 200.0f * (fy - fz);
}

__device__ float ciede2000(float L1, float a1, float b1,
                           float L2, float a2, float b2) {
    const float TINY = 1e-12f;
    const float P7   = 6103515625.0f;       // 25^7
    float C1 = hsqrt(fmaf(a1, a1, fmaf(b1, b1, TINY)));
    float C2 = hsqrt(fmaf(a2, a2, fmaf(b2, b2, TINY)));
    float cb  = 0.5f * (C1 + C2);
    float cb2 = cb * cb;
    float cb7 = cb2 * cb2 * cb2 * cb;       // x^7 with 4 muls
    float G = 0.5f * (1.0f - hsqrt(cb7 * hrcp(cb7 + P7)));
    float a1p = fmaf(G, a1, a1);
    float a2p = fmaf(G, a2, a2);
    float C1p = hsqrt(fmaf(a1p, a1p, fmaf(b1, b1, TINY)));
    float C2p = hsqrt(fmaf(a2p, a2p, fmaf(b2, b2, TINY)));
    float h1p = atan2f(b1, a1p) * 57.29577951308232f; if (h1p < 0.0f) h1p += 360.0f;
    float h2p = atan2f(b2, a2p) * 57.29577951308232f; if (h2p < 0.0f) h2p += 360.0f;

    float dLp = L2 - L1;
    float dCp = C2p - C1p;
    float dh  = h2p - h1p;
    dh = dh > 180.0f ? dh - 360.0f : (dh < -180.0f ? dh + 360.0f : dh);
    // sin(deg2rad(dh)*0.5) -> revolutions: dh/720
    float dHp = 2.0f * hsqrt(C1p * C2p) * sin_rev(dh * (1.0f/720.0f));

    float Lb   = 0.5f * (L1 + L2);
    float Cbp  = 0.5f * (C1p + C2p);
    float hsum = h1p + h2p;
    float hbar = fabsf(h1p - h2p) <= 180.0f ? 0.5f * hsum
               : (hsum < 360.0f ? 0.5f * (hsum + 360.0f) : 0.5f * (hsum - 360.0f));

    float T = 1.0f
            - 0.17f * cosd(hbar - 30.0f)
            + 0.24f * cosd(2.0f * hbar)
            + 0.32f * cosd(fmaf(3.0f, hbar, 6.0f))
            - 0.20f * cosd(fmaf(4.0f, hbar, -63.0f));

    float l50 = Lb - 50.0f;
    float l2  = l50 * l50;
    float SL  = fmaf(0.015f * l2, hrsq(20.0f + l2), 1.0f);
    float SC  = fmaf(0.045f, Cbp, 1.0f);
    float SH  = fmaf(0.015f * Cbp, T, 1.0f);
    float cp2 = Cbp * Cbp;
    float cp7 = cp2 * cp2 * cp2 * Cbp;
    float RC  = 2.0f * hsqrt(cp7 * hrcp(cp7 + P7));
    float hh  = (hbar - 275.0f) * (1.0f/25.0f);
    float dth = 30.0f * hexp2(-1.4426950408889634f * hh * hh); // exp(-hh^2)
    // sin(deg2rad(2*dth)) -> revolutions: dth/180
    float RT  = -sin_rev(dth * (1.0f/180.0f)) * RC;

    float tL = dLp * hrcp(SL);
    float tC = dCp * hrcp(SC);
    float tH = dHp * hrcp(SH);
    float s  = fmaf(tL, tL, fmaf(tC, tC, fmaf(tH, tH, RT * tC * tH)));
    return hsqrt(fmaxf(s, 0.0f));
}

#if HAVE_TDM
// One TDM descriptor: 2D tile, tile_dim0 = TILE f32 elements, tile_dim1 = 3
// channel planes, tensor_dim0_stride = HW elements (NCHW plane stride).
// Lands 3 contiguous rows of TILE floats at ldsOff.
__device__ __forceinline__ void tdm_issue_3ch(const float* g, unsigned ldsOff) {
    unsigned long long ga = (unsigned long long)(uintptr_t)g;
    u32x4 g0;
    g0.x = 1u;                                   // count=1, user load descriptor
    g0.y = ldsOff;                               // LDS byte address
    g0.z = (unsigned)ga;                         // global addr [31:0]
    g0.w = (unsigned)((ga >> 32) & 0x1FFFFFFull) // global addr [56:32]
         | (2u << 30);                           // type = 2 ("image")
    i32x8 g1;
    g1[0] = 0x00020000;                          // data_size=2 (4 bytes)
    g1[1] = (int)(((unsigned)TILE) << 16);       // tensor_dim0[15:0] @ bits 63:48
    g1[2] = (int)(3u << 16);                     // tensor_dim1=3 @ bits 95:80
    g1[3] = (int)(((unsigned)TILE) << 16);       // tile_dim0 @ bits 127:112
    g1[4] = 3;                                   // tile_dim1=3 @ bits 143:128
    g1[5] = HW;                                  // tensor_dim0_stride[31:0]
    g1[6] = 0;
    g1[7] = 0;
    i32x4 z4 = {0, 0, 0, 0};                     // groups 2/3 unused (<=2D)
    i32x8 z8 = {0, 0, 0, 0, 0, 0, 0, 0};         // trailing group (clang-23 form)
    __builtin_amdgcn_tensor_load_to_lds(g0, g1, z4, z4, z8, 0);
}
#endif

__global__ __launch_bounds__(THREADS)
void colorloss_main(const float* __restrict__ x, const float* __restrict__ y,
                    float* __restrict__ partial, int tiles, int gstride) {
    __shared__ float buf[2][6][TILE];   // double-buffered: [x r,g,b | y r,g,b]
    __shared__ float red[THREADS];
    const int tid = threadIdx.x;
    float acc = 0.0f;

#if HAVE_TDM
    // TDM ignores EXEC, so gate issue on a *scalar* (readfirstlane) branch:
    // exactly one wave per workgroup drives the DMA pipeline.
    const bool leader = (__builtin_amdgcn_readfirstlane((int)tid) == 0);
    int t = blockIdx.x;
    int par = 0;
    if (t < tiles && leader) {
        long long sp = (long long)t * TILE;
        int b = (int)(sp >> 18), p = (int)(sp & (HW - 1));
        tdm_issue_3ch(x + ((size_t)b * 3 * HW + p), (unsigned)(uintptr_t)&buf[0][0][0]);
        tdm_issue_3ch(y + ((size_t)b * 3 * HW + p), (unsigned)(uintptr_t)&buf[0][3][0]);
    }
    for (; t < tiles; t += gstride) {
        int tn = t + gstride;
        if (leader) {
            if (tn < tiles) {
                long long sp = (long long)tn * TILE;
                int b = (int)(sp >> 18), p = (int)(sp & (HW - 1));
                tdm_issue_3ch(x + ((size_t)b * 3 * HW + p), (unsigned)(uintptr_t)&buf[par ^ 1][0][0]);
                tdm_issue_3ch(y + ((size_t)b * 3 * HW + p), (unsigned)(uintptr_t)&buf[par ^ 1][3][0]);
                __builtin_amdgcn_s_wait_tensorcnt(2);  // current tile's 2 DMAs done
            } else {
                __builtin_amdgcn_s_wait_tensorcnt(0);  // drain for last tile
            }
        }
        __syncthreads();                                // publish LDS to all waves
        {
            const float (*bp)[TILE] = buf[par];
#pragma unroll
            for (int k = 0; k < TILE / THREADS; ++k) {
                int j = tid + k * THREADS;
                float Lx, Ax, Bx, Ly, Ay, By;
                rgb2lab(bp[0][j], bp[1][j], bp[2][j], Lx, Ax, Bx);
                rgb2lab(bp[3][j], bp[4][j], bp[5][j], Ly, Ay, By);
                acc += fabsf(ciede2000(Lx, Ax, Bx, Ly, Ay, By));
            }
        }
        __syncthreads();   // everyone done reading before this buffer is re-DMA'd
        par ^= 1;
    }
#else
    // Fallback (toolchains without the TDM builtin): cooperative staging.
    for (int t = blockIdx.x; t < tiles; t += gstride) {
        long long sp = (long long)t * TILE;
        int b = (int)(sp >> 18), p = (int)(sp & (HW - 1));
        const float* gx = x + ((size_t)b * 3 * HW + p);
        const float* gy = y + ((size_t)b * 3 * HW + p);
        for (int c = 0; c < 3; ++c)
            for (int j = tid; j < TILE; j += THREADS) {
                buf[0][c][j]     = gx[(size_t)c * HW + j];
                buf[0][c + 3][j] = gy[(size_t)c * HW + j];
            }
        __syncthreads();
#pragma unroll
        for (int k = 0; k < TILE / THREADS; ++k) {
            int j = tid + k * THREADS;
            float Lx, Ax, Bx, Ly, Ay, By;
            rgb2lab(buf[0][0][j], buf[0][1][j], buf[0][2][j], Lx, Ax, Bx);
            rgb2lab(buf[0][3][j], buf[0][4][j], buf[0][5][j], Ly, Ay, By);
            acc += fabsf(ciede2000(Lx, Ax, Bx, Ly, Ay, By));
        }
        __syncthreads();
    }
#endif

    // deterministic block reduction
    red[tid] = acc;
    __syncthreads();
    for (int s = THREADS / 2; s > 0; s >>= 1) {
        if (tid < s) red[tid] += red[tid + s];
        __syncthreads();
    }
    if (tid == 0) partial[blockIdx.x] = red[0];
}

__global__ __launch_bounds__(THREADS)
void colorloss_reduce(const float* __restrict__ partial, int n,
                      float* __restrict__ out, float scale) {
    __shared__ float red[THREADS];
    float s = 0.0f;
    for (int i = threadIdx.x; i < n; i += THREADS) s += partial[i];
    red[threadIdx.x] = s;
    __syncthreads();
    for (int k = THREADS / 2; k > 0; k >>= 1) {
        if (threadIdx.x < k) red[threadIdx.x] += red[threadIdx.x + k];
        __syncthreads();
    }
    if (threadIdx.x == 0) out[0] = red[0] * scale;
}

extern "C" void kernel_launch(void* const* d_in, const int* in_sizes, int n_in,
                              void* d_out, int out_size, void* d_ws, size_t ws_size,
                              hipStream_t stream) {
    (void)n_in; (void)out_size; (void)ws_size;
    const float* x = (const float*)d_in[0];
    const float* y = (const float*)d_in[1];
    float* out     = (float*)d_out;
    float* partial = (float*)d_ws;

    const int total = in_sizes[0];       // 32*3*512*512
    const int npix  = total / 3;         // 8,388,608 pixels
    const int tiles = npix / TILE;       // 8192
    int blocks = tiles < MAXBLOCKS ? tiles : MAXBLOCKS;

    colorloss_main<<<blocks, THREADS, 0, stream>>>(x, y, partial, tiles, blocks);

    const float scale = 1.0f / ((float)npix * 100.0f);  // mean / 100
    colorloss_reduce<<<1, THREADS, 0, stream>>>(partial, blocks, out, scale);
}